// JAGeRLoss_8263517078114
// MI455X (gfx1250) — compile-verified
//
#include <hip/hip_runtime.h>
#include <math.h>

// ---- problem constants (match reference) ----
#define KK      9
#define HH      5
#define MMF     59049.0f            // K^H
#define NSHELL  321                 // H*(K-1)^2 + 1
#define NSHELL4 324                 // padded to multiple of 4 for WMMA K-chunks
#define C_COEF  0.49996397161691486f
#define EPSV    1e-8f
#define HLOGK   10.986122886681098f // H*ln(K)

typedef __attribute__((ext_vector_type(2))) float v2f;
typedef __attribute__((ext_vector_type(8))) float v8f;

#define GP 16     // batch elements per block (one WMMA tile of rows)
#define SP 328    // padded shell stride (8B-aligned pairs, bank spread)

// Kernel 0: copy lam -> out[1..N], rho -> out[1+N..2N]
__global__ void copy_init(const float* __restrict__ lam, const float* __restrict__ rho,
                          float* __restrict__ out, int N) {
    int i = blockIdx.x * blockDim.x + threadIdx.x;
    if (i < N)          out[1 + i] = lam[i];
    else if (i < 2 * N) out[1 + i] = rho[i - N];   // 1+N+(i-N) == 1+i
}

// Kernel 1: fused per-batch computation. 16 batch elems per 256-thread block.
__global__ __launch_bounds__(256) void jager_main(
    const float* __restrict__ logits, const int* __restrict__ idx,
    const int* __restrict__ Y, const float* __restrict__ lam,
    const float* __restrict__ rho, float* __restrict__ out_lam,
    float* __restrict__ out_rho, float* __restrict__ ws_loss, int B)
{
    __shared__ __align__(16) float shA[GP * SP];   // conv ping (final p_shell)
    __shared__ __align__(16) float shB[GP * SP];   // conv pong
    __shared__ float shPr[GP][HH][KK];             // per-digit softmax probs
    __shared__ int   shY[GP][HH];
    __shared__ float shKpi[GP], shRho[GP], shSrho[GP], shLam[GP], shNll[GP];
    __shared__ float shT[16][2];                   // WMMA result: t1, t2 per row b

    const int t   = threadIdx.x;
    const int g   = t >> 4;                 // group (batch elem within block)
    const int sub = t & 15;                 // lane within group
    const int b   = blockIdx.x * GP + g;
    const bool valid = (b < B);

    // ---------- Phase 0: per-b scalars, softmax, Z_mode ----------
    if (sub == 0 && valid) {
        int ib = idx[b];
        float logp_t = 0.f, logZmode = 0.f;
        for (int h = 0; h < HH; ++h) {
            int y = Y[ib * HH + h];
            shY[g][h] = y;
            float l[KK], mx = -1e30f;
            for (int k = 0; k < KK; ++k) { l[k] = logits[(b * HH + h) * KK + k]; mx = fmaxf(mx, l[k]); }
            float se = 0.f;
            for (int k = 0; k < KK; ++k) se += expf(l[k] - mx);
            float lse = mx + logf(se);
            for (int k = 0; k < KK; ++k) shPr[g][h][k] = expf(l[k] - lse);
            logp_t += l[y] - lse;
            float z = 0.f;
            for (int j = 0; j < KK; ++j) { float d = (float)(y - j); z += expf(-C_COEF * d * d); }
            logZmode += logf(z);
        }
        float Kpi_t = MMF * expf(-logZmode);          // M / Z_mode
        float rb = rho[ib], lb = lam[ib];
        float S_t = 1.f + rb * (Kpi_t - 1.f);
        float nll = -(logp_t + logf(S_t + EPSV));
        shKpi[g] = Kpi_t; shRho[g] = rb; shSrho[g] = (rb == 0.f) ? 1.f : rb;
        shLam[g] = lb; shNll[g] = nll;
        ws_loss[b] = lb * nll;                        // loss summed deterministically later
    }
    // zero conv buffer including the 3 pad shells (321..323) used by the WMMA tail chunk
    for (int s = sub; s < NSHELL4; s += 16) shA[g * SP + s] = 0.f;
    __syncthreads();

    // ---------- Phase 1: 5-fold shell convolution (replaces 59049-wide segment_sum) ----------
    if (sub == 0) {
        int y0 = shY[g][0];
        for (int k = 0; k < KK; ++k) {
            int d = (k - y0) * (k - y0);
            shA[g * SP + d] += shPr[g][0][k];
        }
    }
    __syncthreads();
    float* cur = shA; float* nxt = shB;
    for (int h = 1; h < HH; ++h) {
        int y = shY[g][h];
        for (int s = sub; s < NSHELL; s += 16) {
            float acc = 0.f;
            #pragma unroll
            for (int k = 0; k < KK; ++k) {
                int d = (k - y) * (k - y);
                int sp = s - d;
                if (sp >= 0) acc += shPr[g][h][k] * cur[g * SP + sp];
            }
            nxt[g * SP + s] = acc;
        }
        __syncthreads();
        float* tmp = cur; cur = nxt; nxt = tmp;
    }
    // After 4 conv steps the final p_shell is in shA; shA[321..323] still zero
    // (steps write only s<321, and step parity lands the result back in shA).

    // ---------- Phase 2: WMMA f32 16x16x4 GEMM  T[16x2] = P_shell[16x324] x ker^{1,2}[324x2] ----------
    if (t < 32) {                      // whole wave 0 -> EXEC all ones at the WMMA
        const int lane = t;
        const int row  = lane & 15;    // A-matrix row (batch elem in block)
        const int hi   = lane >> 4;    // K-half select per ISA A layout
        const int col  = lane & 15;    // B/D column
        const int koff = hi * 2;       // this lane's K offset within a 4-chunk
        // Geometric weight recurrence: bb.x = e^{-C*(col+1)*sx}, sx = 4c+koff.
        const float o   = (float)(col + 1);
        const bool  act = (col < 2);
        float wx  = act ? expf(-C_COEF * o * (float)koff) : 0.f;  // value at c=0
        const float eo  = act ? expf(-C_COEF * o)         : 0.f;  // step s -> s+1
        const float f4  = act ? expf(-4.f * C_COEF * o)   : 0.f;  // step c -> c+1
        const float* pa = &shA[row * SP + koff];                   // 8B-aligned
        v8f acc = {};
        for (int c = 0; c < 81; ++c) {
            v2f a = *(const v2f*)(pa + 4 * c);    // single ds_load_b64
            v2f bb; bb.x = wx; bb.y = wx * eo;
            acc = __builtin_amdgcn_wmma_f32_16x16x4_f32(false, a, false, bb,
                                                        (short)0, acc, false, false);
            wx *= f4;
        }
        #pragma unroll
        for (int v = 0; v < 8; ++v) {
            int Mrow = v + hi * 8;     // D layout: VGPR v -> M=v (lo half) / v+8 (hi half)
            if (col < 2) shT[Mrow][col] = acc[v];
        }
    }
    __syncthreads();

    // ---------- Phase 3: per-b moments (rho>0 branch), update rho/lam, scatter ----------
    {
        float Kpi_t = shKpi[g];
        float srho  = shSrho[g];
        float m1 = 0.f, m2 = 0.f;
        float w = Kpi_t * expf(-C_COEF * (float)sub);   // Kpi_shell at s=sub
        const float step = expf(-16.f * C_COEF);        // s -> s+16
        for (int s = sub; s < NSHELL; s += 16) {
            float p   = shA[g * SP + s];
            float km1 = w - 1.f;
            float S   = 1.f + srho * km1;
            float ratio = km1 / S;     // == -expm1(-log1p(srho*km1))/srho
            m1 += p * ratio;
            m2 += p * ratio * ratio;
            w  *= step;
        }
        // reduce across the 16 contiguous lanes of this group (wave32)
        for (int off = 8; off >= 1; off >>= 1) {
            m1 += __shfl_xor(m1, off, 16);
            m2 += __shfl_xor(m2, off, 16);
        }
        if (sub == 0 && valid) {
            float rb = shRho[g], lb = shLam[g], nll = shNll[g];
            float t1 = shT[g][0], t2 = shT[g][1];       // from WMMA
            float k1 = Kpi_t * t1;
            float k2 = Kpi_t * Kpi_t * t2;
            float d0 = (rb == 0.f) ? (k1 - 1.f)              : m1;
            float d1 = (rb == 0.f) ? (k2 - 2.f * k1 + 1.f)   : m2;
            float rho_new = rb - d0 / (d1 + EPSV);
            rho_new = fminf(fmaxf(rho_new, 0.f), 1.f);
            float lam_tgt = HLOGK / (nll + EPSV);           // LAM0 == 1
            float lam_new = 0.9f * lb + 0.1f * fminf(lam_tgt, 1.0f);
            lam_new = fminf(fmaxf(lam_new, 0.5f), 1.0f);    // [LAM0/ALPHA, LAM0]
            int ib = idx[b];
            out_lam[ib] = lam_new;
            out_rho[ib] = rho_new;
        }
    }
}

// Kernel 2: deterministic loss reduction (fixed tree order -> replay-stable)
__global__ void loss_reduce(const float* __restrict__ ws, float* __restrict__ out, int B) {
    __shared__ float red[256];
    int t = threadIdx.x;
    float v = 0.f;
    for (int i = t; i < B; i += 256) v += ws[i];
    red[t] = v; __syncthreads();
    for (int s = 128; s > 0; s >>= 1) { if (t < s) red[t] += red[t + s]; __syncthreads(); }
    if (t == 0) out[0] = red[0] / (float)B;
}

extern "C" void kernel_launch(void* const* d_in, const int* in_sizes, int n_in,
                              void* d_out, int out_size, void* d_ws, size_t ws_size,
                              hipStream_t stream) {
    const float* logits = (const float*)d_in[0];
    const int*   idx    = (const int*)d_in[1];
    const int*   Y      = (const int*)d_in[2];
    const float* lam    = (const float*)d_in[3];
    const float* rho    = (const float*)d_in[4];
    const int B = in_sizes[1];
    const int N = in_sizes[3];
    float* out = (float*)d_out;
    float* ws  = (float*)d_ws;      // B floats of per-sample loss terms

    int total = 2 * N;
    copy_init<<<(total + 255) / 256, 256, 0, stream>>>(lam, rho, out, N);
    jager_main<<<(B + GP - 1) / GP, 256, 0, stream>>>(logits, idx, Y, lam, rho,
                                                      out + 1, out + 1 + N, ws, B);
    loss_reduce<<<1, 256, 0, stream>>>(ws, out, B);
}